// LReLu_torch_4234837754061
// MI455X (gfx1250) — compile-verified
//
#include <hip/hip_runtime.h>

typedef float v2f __attribute__((ext_vector_type(2)));
typedef float v8f __attribute__((ext_vector_type(8)));
typedef int   v4i __attribute__((ext_vector_type(4)));

#if __has_builtin(__builtin_amdgcn_global_load_async_to_lds_b128) && \
    __has_builtin(__builtin_amdgcn_s_wait_asynccnt)
#define USE_ASYNC_LDS 1
typedef __attribute__((address_space(1))) v4i* g_v4i_p;   // global (prints as __device__)
typedef __attribute__((address_space(3))) v4i* l_v4i_p;   // LDS
#endif

// ---------------------------------------------------------------------------
// Keys cubic kernel, a = -0.5 (matches jax.image 'cubic' / torch bicubic)
// ---------------------------------------------------------------------------
__device__ __forceinline__ float cubic_keys(float x) {
    x = fabsf(x);
    if (x < 1.f) return (1.5f * x - 2.5f) * x * x + 1.f;
    if (x < 2.f) return ((-0.5f * x + 2.5f) * x - 4.f) * x + 2.f;
    return 0.f;
}

// Build resize weight matrix W[out][in] (jax.image.resize semantics):
//   inv_scale = in/out ; ks = max(inv_scale, 1)  (antialias)
//   sample_f(o) = (o+0.5)*inv_scale - 0.5
//   W[o][i] = k((sample_f-i)/ks) / sum_i k(...)
__global__ void build_resize_matrix(float* __restrict__ W, int in, int out) {
    int o = blockIdx.x * blockDim.x + threadIdx.x;
    if (o >= out) return;
    float inv_scale = (float)in / (float)out;
    float ks = inv_scale > 1.f ? inv_scale : 1.f;
    float inv_ks = 1.f / ks;
    float sf = ((float)o + 0.5f) * inv_scale - 0.5f;
    float sum = 0.f;
    for (int i = 0; i < in; ++i) sum += cubic_keys((sf - (float)i) * inv_ks);
    float inv = 1.f / sum;
    for (int i = 0; i < in; ++i)
        W[o * in + i] = cubic_keys((sf - (float)i) * inv_ks) * inv;
}

// Ut[i][o] = U[o][i]   (U: 256x128  ->  Ut: 128x256)
__global__ void transpose_u(const float* __restrict__ U, float* __restrict__ Ut) {
    int idx = blockIdx.x * blockDim.x + threadIdx.x;   // 256*128
    if (idx >= 256 * 128) return;
    int o = idx >> 7, i = idx & 127;
    Ut[i * 256 + o] = U[o * 128 + i];
}

// D = D2 @ D1 : (64x128)@(128x256) -> 64x256 ; also Dt = D^T (256x64)
__global__ void compose_d(const float* __restrict__ D2, const float* __restrict__ D1,
                          float* __restrict__ D, float* __restrict__ Dt) {
    int idx = blockIdx.x * blockDim.x + threadIdx.x;   // 64*256
    if (idx >= 64 * 256) return;
    int o = idx >> 8, i = idx & 255;
    float s = 0.f;
    for (int j = 0; j < 128; ++j) s += D2[o * 128 + j] * D1[j * 256 + i];
    D[o * 256 + i] = s;
    Dt[i * 64 + o] = s;
}

// ---------------------------------------------------------------------------
// Fused per-plane kernel:  out = D @ relu(U @ X @ U^T) @ D^T + bias
// One workgroup (256 threads = 8 wave32) per (n,c) plane.
// ---------------------------------------------------------------------------
#define XS_STRIDE 132   // 128 + pad (bank-conflict avoidance, keeps 16B align)
#define MS_STRIDE 132
#define TS_STRIDE 260
#define ZS_STRIDE 260

__device__ __forceinline__ v8f wmma_f32(v2f a, v2f b, v8f c) {
    return __builtin_amdgcn_wmma_f32_16x16x4_f32(
        /*neg_a=*/false, a, /*neg_b=*/false, b,
        /*c_mod=*/(short)0, c, /*reuse_a=*/false, /*reuse_b=*/false);
}

__global__ __launch_bounds__(256)
void fused_biresize_kernel(const float* __restrict__ x, const float* __restrict__ bias,
                           const float* __restrict__ U, const float* __restrict__ Ut,
                           const float* __restrict__ D, const float* __restrict__ Dt,
                           float* __restrict__ out) {
    __shared__ float Xs[128 * XS_STRIDE];   // input plane 128x128
    __shared__ float Ms[16 * MS_STRIDE];    // M  = U_bi @ X      (16x128)
    __shared__ float Ts[16 * TS_STRIDE];    // T  = relu(M @ U^T) (16x256)
    __shared__ float Zs[64 * ZS_STRIDE];    // Z  = D @ relu(T)   (64x256)

    const int tid  = threadIdx.x;
    const int wave = tid >> 5;          // 0..7
    const int lane = tid & 31;
    const int half = lane >> 4;         // 0/1
    const int ln   = lane & 15;         // 0..15
    const int plane = blockIdx.x;       // n*128 + c
    const float* __restrict__ xp = x + (size_t)plane * (128 * 128);

    // ---- stage X into LDS --------------------------------------------------
#ifdef USE_ASYNC_LDS
    // CDNA5 async copy: 16B per lane per op, no VGPR round-trip (ASYNCcnt).
    for (int i = tid; i < (128 * 128) / 4; i += 256) {
        int e = i * 4;
        int r = e >> 7, c = e & 127;
        __builtin_amdgcn_global_load_async_to_lds_b128(
            (g_v4i_p)((float*)xp + e),
            (l_v4i_p)&Xs[r * XS_STRIDE + c],
            0, 0);
    }
    __builtin_amdgcn_s_wait_asynccnt(0);   // each wave waits its own async ops
    __syncthreads();                       // then all waves' data is visible
#else
    for (int i = tid; i < (128 * 128) / 4; i += 256) {
        int e = i * 4;
        int r = e >> 7, c = e & 127;
        float4 v = ((const float4*)xp)[i];
        *(float4*)&Xs[r * XS_STRIDE + c] = v;
    }
    __syncthreads();
#endif

    // ---- Z accumulators: wave w owns Z[0:64, w*32 : w*32+32] in registers --
    v8f zacc[4][2];
    for (int m = 0; m < 4; ++m)
        for (int t = 0; t < 2; ++t)
            for (int r = 0; r < 8; ++r) zacc[m][t][r] = 0.f;

    for (int bi = 0; bi < 16; ++bi) {
        // ---- step a: M = U[bi*16:+16, :] @ X  (16x128); wave w -> cols w*16..+16
        {
            v8f acc; for (int r = 0; r < 8; ++r) acc[r] = 0.f;
            const float* __restrict__ arow = U + (size_t)(bi * 16 + ln) * 128 + 2 * half;
            const int ncol = wave * 16 + ln;
#pragma unroll 8
            for (int k = 0; k < 32; ++k) {
                v2f a; a.x = arow[4 * k]; a.y = arow[4 * k + 1];
                int krow = 4 * k + 2 * half;
                v2f b;
                b.x = Xs[krow * XS_STRIDE + ncol];
                b.y = Xs[(krow + 1) * XS_STRIDE + ncol];
                acc = wmma_f32(a, b, acc);
            }
            for (int r = 0; r < 8; ++r)
                Ms[(r + 8 * half) * MS_STRIDE + ncol] = acc[r];
        }
        __syncthreads();

        // ---- step b: T = relu(M @ Ut)  (16x256); wave w -> cols w*32..+32
        // Both n-tiles of a wave share the A fragment (rows of M).
        {
            const int ncol0 = wave * 32 + ln;
            const int ncol1 = ncol0 + 16;
            v8f acc0, acc1;
            for (int r = 0; r < 8; ++r) { acc0[r] = 0.f; acc1[r] = 0.f; }
            const float* __restrict__ arow = &Ms[ln * MS_STRIDE + 2 * half];
#pragma unroll 8
            for (int k = 0; k < 32; ++k) {
                v2f a; a.x = arow[4 * k]; a.y = arow[4 * k + 1];
                int krow = 4 * k + 2 * half;
                v2f b0, b1;
                b0.x = Ut[(size_t)krow * 256 + ncol0];
                b0.y = Ut[(size_t)(krow + 1) * 256 + ncol0];
                b1.x = Ut[(size_t)krow * 256 + ncol1];
                b1.y = Ut[(size_t)(krow + 1) * 256 + ncol1];
                acc0 = wmma_f32(a, b0, acc0);
                acc1 = wmma_f32(a, b1, acc1);
            }
            for (int r = 0; r < 8; ++r) {
                float v0 = acc0[r], v1 = acc1[r];
                v0 = v0 > 0.f ? v0 : 0.01f * v0;             // LeakyReLU(0.01)
                v1 = v1 > 0.f ? v1 : 0.01f * v1;
                Ts[(r + 8 * half) * TS_STRIDE + ncol0] = v0;
                Ts[(r + 8 * half) * TS_STRIDE + ncol1] = v1;
            }
        }
        __syncthreads();

        // ---- step d: Z += D[:, bi*16:+16] @ T  (64x256, in registers)
        // Hoist B fragments (shared across m) and A fragments (shared across t).
        {
            v2f bt[2][4];
#pragma unroll
            for (int t = 0; t < 2; ++t) {
                const int ncol = wave * 32 + t * 16 + ln;
#pragma unroll
                for (int k = 0; k < 4; ++k) {
                    int krow = 4 * k + 2 * half;
                    bt[t][k].x = Ts[krow * TS_STRIDE + ncol];
                    bt[t][k].y = Ts[(krow + 1) * TS_STRIDE + ncol];
                }
            }
#pragma unroll
            for (int m = 0; m < 4; ++m) {
                const float* __restrict__ arow =
                    D + (size_t)(m * 16 + ln) * 256 + bi * 16 + 2 * half;
                v2f am[4];
#pragma unroll
                for (int k = 0; k < 4; ++k) { am[k].x = arow[4 * k]; am[k].y = arow[4 * k + 1]; }
#pragma unroll
                for (int t = 0; t < 2; ++t) {
                    v8f acc = zacc[m][t];
#pragma unroll
                    for (int k = 0; k < 4; ++k) acc = wmma_f32(am[k], bt[t][k], acc);
                    zacc[m][t] = acc;
                }
            }
        }
        __syncthreads();   // protect Ms/Ts for next bi iteration
    }

    // ---- dump Z to LDS -----------------------------------------------------
    for (int m = 0; m < 4; ++m)
        for (int t = 0; t < 2; ++t)
            for (int r = 0; r < 8; ++r)
                Zs[(m * 16 + r + 8 * half) * ZS_STRIDE + wave * 32 + t * 16 + ln] =
                    zacc[m][t][r];
    __syncthreads();

    // ---- epilogue: out = Z @ Dt + bias  (64x64); 16 tiles, 2 per wave ------
    // Wave w's two tiles are (mT, nT0) and (mT, nT0+1): A fragment shared.
    const float bv = bias[plane & 127];
    float* __restrict__ outp = out + (size_t)plane * (64 * 64);
    {
        const int mT  = wave >> 1;
        const int nT0 = (wave & 1) * 2;
        const int nc0 = nT0 * 16 + ln;
        const int nc1 = nc0 + 16;
        v8f acc0, acc1;
        for (int r = 0; r < 8; ++r) { acc0[r] = 0.f; acc1[r] = 0.f; }
        const float* __restrict__ arow = &Zs[(mT * 16 + ln) * ZS_STRIDE + 2 * half];
#pragma unroll 8
        for (int k = 0; k < 64; ++k) {
            v2f a; a.x = arow[4 * k]; a.y = arow[4 * k + 1];
            int krow = 4 * k + 2 * half;
            v2f b0, b1;
            b0.x = Dt[(size_t)krow * 64 + nc0];
            b0.y = Dt[(size_t)(krow + 1) * 64 + nc0];
            b1.x = Dt[(size_t)krow * 64 + nc1];
            b1.y = Dt[(size_t)(krow + 1) * 64 + nc1];
            acc0 = wmma_f32(a, b0, acc0);
            acc1 = wmma_f32(a, b1, acc1);
        }
        for (int r = 0; r < 8; ++r) {
            outp[(mT * 16 + r + 8 * half) * 64 + nc0] = acc0[r] + bv;
            outp[(mT * 16 + r + 8 * half) * 64 + nc1] = acc1[r] + bv;
        }
    }
}

// ---------------------------------------------------------------------------
// Launch
// ---------------------------------------------------------------------------
extern "C" void kernel_launch(void* const* d_in, const int* in_sizes, int n_in,
                              void* d_out, int out_size, void* d_ws, size_t ws_size,
                              hipStream_t stream) {
    const float* x    = (const float*)d_in[0];   // (8,128,128,128) f32
    const float* bias = (const float*)d_in[1];   // (128,) f32
    float* out = (float*)d_out;                  // (8,128,64,64) f32

    // workspace layout (floats)
    float* U  = (float*)d_ws;          // 256*128
    float* Ut = U  + 256 * 128;        // 128*256
    float* D1 = Ut + 128 * 256;        // 128*256  (256->128 matrix)
    float* D2 = D1 + 128 * 256;        // 64*128   (128->64 matrix)
    float* D  = D2 + 64 * 128;         // 64*256
    float* Dt = D  + 64 * 256;         // 256*64
    (void)ws_size; (void)in_sizes; (void)n_in; (void)out_size;

    // Build resize operators (recomputed every call: deterministic, ~0.5 MB)
    build_resize_matrix<<<4, 64, 0, stream>>>(U, 128, 256);   // upsample 2x
    build_resize_matrix<<<2, 64, 0, stream>>>(D1, 256, 128);  // down 2x (AA)
    build_resize_matrix<<<1, 64, 0, stream>>>(D2, 128, 64);   // down 2x (AA)
    transpose_u<<<128, 256, 0, stream>>>(U, Ut);
    compose_d<<<64, 256, 0, stream>>>(D2, D1, D, Dt);

    // Fused main kernel: one workgroup per (n,c) plane
    fused_biresize_kernel<<<8 * 128, 256, 0, stream>>>(x, bias, U, Ut, D, Dt, out);
}